// _ConvBlock7_43018392436812
// MI455X (gfx1250) — compile-verified
//
#include <hip/hip_runtime.h>
#include <hip/hip_bf16.h>

typedef __attribute__((ext_vector_type(2))) float v2f;
typedef __attribute__((ext_vector_type(8))) float v8f;

#define CCH 64
#define PAIR_STRIDE 160  // 128 payload floats + 32 pad: half-waves hit disjoint LDS bank halves

// ---------------------------------------------------------------------------
// zero fill
// ---------------------------------------------------------------------------
__global__ void zero_f32(float* __restrict__ p, long n) {
  long i = (long)blockIdx.x * blockDim.x + threadIdx.x;
  if (i < n) p[i] = 0.0f;
}

// ---------------------------------------------------------------------------
// edge-weighted gather + scatter-sum:  agg[dst] += x[src] * w   (C=64 channels)
// one thread handles one edge's 4-channel group (float4 gather, 4 f32 atomics)
// also reused for the unpooling stage (same math, different index arrays).
// ---------------------------------------------------------------------------
__global__ void edge_scatter(const float* __restrict__ x,
                             const int* __restrict__ src_idx,
                             const int* __restrict__ dst_idx,
                             const float* __restrict__ w,
                             float* __restrict__ agg, int E) {
  int t = blockIdx.x * blockDim.x + threadIdx.x;
  int e = t >> 4;
  if (e >= E) return;
  int cg = (t & 15) << 2;
  int s = src_idx[e];
  int d = dst_idx[e];
  float ww = w[e];
  const float4 xv = *(const float4*)(x + (long)s * CCH + cg);
  float* p = agg + (long)d * CCH + cg;
  __hip_atomic_fetch_add(p + 0, xv.x * ww, __ATOMIC_RELAXED, __HIP_MEMORY_SCOPE_AGENT);
  __hip_atomic_fetch_add(p + 1, xv.y * ww, __ATOMIC_RELAXED, __HIP_MEMORY_SCOPE_AGENT);
  __hip_atomic_fetch_add(p + 2, xv.z * ww, __ATOMIC_RELAXED, __HIP_MEMORY_SCOPE_AGENT);
  __hip_atomic_fetch_add(p + 3, xv.w * ww, __ATOMIC_RELAXED, __HIP_MEMORY_SCOPE_AGENT);
}

// ---------------------------------------------------------------------------
// Fused dual GEMM via f32 WMMA:
//   out = Aagg @ Wn + Aroot @ Wr + bias       (all [*,64] / [64,64])
// Treated as one K=128 GEMM: A_cat = [Aagg | Aroot], W_cat = [Wn ; Wr].
// One wave computes a 16-row tile across all 4 column tiles (N=64).
// W_cat staged in LDS in K-pair-interleaved layout matching the
// V_WMMA_F32_16X16X4_F32 B-fragment layout.
// ---------------------------------------------------------------------------
__global__ __launch_bounds__(256)
void gemm_dual_wmma(const float* __restrict__ Aagg,
                    const float* __restrict__ Aroot,
                    const float* __restrict__ Wn,
                    const float* __restrict__ Wr,
                    const float* __restrict__ bias,
                    float* __restrict__ out, int Nrows) {
  __shared__ float lds[64 * PAIR_STRIDE];

  // Stage W_cat[k][n]: k<64 -> Wn, k>=64 -> Wr, at lds[(k>>1)*PS + 2n + (k&1)]
  for (int idx = threadIdx.x; idx < 128 * 64; idx += 256) {
    int k = idx >> 6, n = idx & 63;
    float v = (k < 64) ? Wn[idx] : Wr[idx - 4096];
    lds[(k >> 1) * PAIR_STRIDE + (n << 1) + (k & 1)] = v;
  }
  __syncthreads();

  const int wave = threadIdx.x >> 5;
  const int lane = threadIdx.x & 31;
  const int lh = lane >> 4;   // half-wave select (K offset +2 / M offset +8)
  const int l  = lane & 15;

  long tile = (long)blockIdx.x * 8 + wave;
  long row0 = tile * 16;
  if (row0 >= Nrows) return;          // wave-uniform exit; EXEC stays all-1s for WMMA

  long arow = row0 + l;
  if (arow >= Nrows) arow = Nrows - 1;  // defensive clamp (Nrows%16==0 in practice)
  const float* pa = Aagg + arow * CCH;
  const float* px = Aroot + arow * CCH;

  v8f acc0 = {}, acc1 = {}, acc2 = {}, acc3 = {};

#pragma unroll
  for (int k0 = 0; k0 < 128; k0 += 4) {
    // A fragment: lane half h holds A[row][k0+2h], A[row][k0+2h+1]
    int kk = k0 + (lh << 1);
    v2f a = (k0 < 64) ? *(const v2f*)(pa + kk)
                      : *(const v2f*)(px + (kk - 64));
    // B fragment: lane half h holds W[k0+2h][n], W[k0+2h+1][n]
    const float* bb = lds + (((k0 >> 1) + lh) * PAIR_STRIDE) + (l << 1);
    v2f b0 = *(const v2f*)(bb);        // cols  0..15
    v2f b1 = *(const v2f*)(bb + 32);   // cols 16..31
    v2f b2 = *(const v2f*)(bb + 64);   // cols 32..47
    v2f b3 = *(const v2f*)(bb + 96);   // cols 48..63
    acc0 = __builtin_amdgcn_wmma_f32_16x16x4_f32(false, a, false, b0, (short)0, acc0, false, false);
    acc1 = __builtin_amdgcn_wmma_f32_16x16x4_f32(false, a, false, b1, (short)0, acc1, false, false);
    acc2 = __builtin_amdgcn_wmma_f32_16x16x4_f32(false, a, false, b2, (short)0, acc2, false, false);
    acc3 = __builtin_amdgcn_wmma_f32_16x16x4_f32(false, a, false, b3, (short)0, acc3, false, false);
  }

  float bc0 = bias[l], bc1 = bias[16 + l], bc2 = bias[32 + l], bc3 = bias[48 + l];

  // D layout: VGPR r -> row r (lanes 0-15) / row r+8 (lanes 16-31), col = lane&15
#pragma unroll
  for (int r = 0; r < 8; ++r) {
    long orow = row0 + r + (lh << 3);
    if (orow < Nrows) {
      float* po = out + orow * CCH + l;
      po[0]  = acc0[r] + bc0;
      po[16] = acc1[r] + bc1;
      po[32] = acc2[r] + bc2;
      po[48] = acc3[r] + bc3;
    }
  }
}

// ---------------------------------------------------------------------------
extern "C" void kernel_launch(void* const* d_in, const int* in_sizes, int n_in,
                              void* d_out, int out_size, void* d_ws, size_t ws_size,
                              hipStream_t stream) {
  const float* x   = (const float*)d_in[0];
  const int*   ei  = (const int*)d_in[1];   // [2,E] flat: [0..E)=src, [E..2E)=dst
  const float* ea  = (const float*)d_in[2];
  const int*   ps  = (const int*)d_in[3];
  const int*   pd  = (const int*)d_in[4];
  const float* pw  = (const float*)d_in[5];
  // d_in[6] = n_fine scalar (derived from out_size instead)
  const float* Wr1 = (const float*)d_in[7];
  const float* Wn1 = (const float*)d_in[8];
  const float* b1  = (const float*)d_in[9];
  const float* Wr2 = (const float*)d_in[10];
  const float* Wn2 = (const float*)d_in[11];
  const float* b2  = (const float*)d_in[12];
  float* out = (float*)d_out;

  const int N  = in_sizes[0] / CCH;
  const int E  = in_sizes[1] / 2;
  const int EP = in_sizes[3];
  const long NC = (long)N * CCH;

  float* agg = (float*)d_ws;      // 25.6 MB
  float* h1  = agg + NC;          // 25.6 MB
  float* h2  = h1 + NC;           // 25.6 MB

  const int* esrc = ei;
  const int* edst = ei + E;

  const int zb  = (int)((NC + 255) / 256);
  const int sb  = (int)(((long)E * 16 + 255) / 256);
  const int pb  = (int)(((long)EP * 16 + 255) / 256);
  const int gb  = ((N + 15) / 16 + 7) / 8;
  const long outN = (long)out_size;
  const int ob  = (int)((outN + 255) / 256);

  // layer 1
  zero_f32<<<zb, 256, 0, stream>>>(agg, NC);
  edge_scatter<<<sb, 256, 0, stream>>>(x, esrc, edst, ea, agg, E);
  gemm_dual_wmma<<<gb, 256, 0, stream>>>(agg, x, Wn1, Wr1, b1, h1, N);
  // layer 2
  zero_f32<<<zb, 256, 0, stream>>>(agg, NC);
  edge_scatter<<<sb, 256, 0, stream>>>(h1, esrc, edst, ea, agg, E);
  gemm_dual_wmma<<<gb, 256, 0, stream>>>(agg, h1, Wn2, Wr2, b2, h2, N);
  // unpool (same gather-scale-scatter pattern)
  zero_f32<<<ob, 256, 0, stream>>>(out, outN);
  edge_scatter<<<pb, 256, 0, stream>>>(h2, ps, pd, pw, out, EP);
}